// CaNet_2602750181783
// MI455X (gfx1250) — compile-verified
//
#include <hip/hip_runtime.h>

#define N_NODES 50000
#define N_EDGES 1000000
#define NTOT    1050000    /* N_EDGES + N_NODES self-loops */
#define IN_F    128
#define OUT_F   128
#define KH      4
#define NEG_SLOPE 0.01f
#define EPS_F   1e-8f

typedef float v2f __attribute__((ext_vector_type(2)));
typedef float v4f __attribute__((ext_vector_type(4)));
typedef float v8f __attribute__((ext_vector_type(8)));

// ---- ordered-uint encoding for float atomicMax (handles negatives) ----
__device__ __forceinline__ unsigned fenc(float f) {
    unsigned u = __float_as_uint(f);
    return (u & 0x80000000u) ? ~u : (u | 0x80000000u);
}
__device__ __forceinline__ float fdec(unsigned u) {
    return (u & 0x80000000u) ? __uint_as_float(u & 0x7FFFFFFFu)
                             : __uint_as_float(~u);
}

// ---------------------------------------------------------------------------
// init: zero per-node degree counters and the 4 per-head max slots
// ---------------------------------------------------------------------------
__global__ void k_init(unsigned* __restrict__ hmax, int* __restrict__ rowcnt) {
    int i = blockIdx.x * blockDim.x + threadIdx.x;
    if (i < N_NODES) rowcnt[i] = 0;
    if (i < KH)      hmax[i]   = 0u;   // below fenc() of any real float
}

// ---------------------------------------------------------------------------
// h[k] = x @ W[k]  via V_WMMA_F32_16X16X4_F32 (fp32 matrix pipe, wave32)
// grid = (N_NODES/16, KH), block = 256 (8 waves); wave w -> col tile w.
// ISA layouts (7.12.2): A 16x4: lane = M (mod 16), vgpr pair = {K0,K1} lanes0-15
//                       and {K2,K3} lanes16-31. B 4x16 symmetric. C/D: 8 vgprs,
//                       vgpr i -> M = i + 8*(lane>=16), N = lane&15.
// ---------------------------------------------------------------------------
__global__ void __launch_bounds__(256) k_gemm(const float* __restrict__ x,
                                              const float* __restrict__ W,
                                              float* __restrict__ h) {
    const int lane = threadIdx.x & 31;
    const int wave = threadIdx.x >> 5;
    const int head = blockIdx.y;
    const int row0 = blockIdx.x << 4;     // 3125 * 16 == 50000 exactly, no tail
    const int n0   = wave << 4;
    const int half = lane >> 4;
    const int m    = lane & 15;

    const float* Wk   = W + (size_t)head * IN_F * OUT_F;
    const float* xrow = x + (size_t)(row0 + m) * IN_F;

    v8f c = {};
#pragma unroll
    for (int k0 = 0; k0 < IN_F; k0 += 4) {
        const int kk = k0 + 2 * half;
        v2f a = *(const v2f*)(xrow + kk);                 // A[M=m][kk], A[M=m][kk+1]
        v2f b;
        b.x = Wk[(size_t)kk       * OUT_F + n0 + m];      // B[kk  ][N=m]
        b.y = Wk[(size_t)(kk + 1) * OUT_F + n0 + m];      // B[kk+1][N=m]
        c = __builtin_amdgcn_wmma_f32_16x16x4_f32(
                /*neg_a=*/false, a, /*neg_b=*/false, b,
                /*c_mod=*/(short)0, c, /*reuse_a=*/false, /*reuse_b=*/false);
    }

    float* hk = h + (size_t)head * N_NODES * OUT_F;
#pragma unroll
    for (int i = 0; i < 8; ++i) {
        const int row = row0 + i + 8 * half;
        hk[(size_t)row * OUT_F + n0 + m] = c[i];
    }
}

// ---------------------------------------------------------------------------
// s_src[k][n] = h[k][n]·a_src[k];  s_dst[k][n] = h[k][n]·a_dst[k]
// one wave per (k, n); shuffle reduction across 32 lanes (wave32).
// ---------------------------------------------------------------------------
__global__ void __launch_bounds__(256) k_scores(const float* __restrict__ h,
                                                const float* __restrict__ a,
                                                float* __restrict__ ssrc,
                                                float* __restrict__ sdst) {
    const int lane = threadIdx.x & 31;
    const int wave = threadIdx.x >> 5;
    const int id   = blockIdx.x * 8 + wave;          // id = k*N_NODES + n
    if (id >= KH * N_NODES) return;
    const int k = id / N_NODES;

    const v4f hv = *(const v4f*)(h + (size_t)id * OUT_F + lane * 4);
    const v4f as = *(const v4f*)(a + (size_t)k * 2 * OUT_F + lane * 4);
    const v4f ad = *(const v4f*)(a + (size_t)k * 2 * OUT_F + OUT_F + lane * 4);

    float s0 = hv.x * as.x + hv.y * as.y + hv.z * as.z + hv.w * as.w;
    float s1 = hv.x * ad.x + hv.y * ad.y + hv.z * ad.z + hv.w * ad.w;
#pragma unroll
    for (int o = 16; o; o >>= 1) {
        s0 += __shfl_xor(s0, o, 32);
        s1 += __shfl_xor(s1, o, 32);
    }
    if (lane == 0) { ssrc[id] = s0; sdst[id] = s1; }
}

// ---------------------------------------------------------------------------
// per-edge: count degree(src) and reduce global per-head max of leakyrelu logit
// ---------------------------------------------------------------------------
__global__ void k_edge(const int* __restrict__ erow, const int* __restrict__ ecol,
                       const float* __restrict__ ssrc, const float* __restrict__ sdst,
                       unsigned* __restrict__ hmax, int* __restrict__ rowcnt) {
    const int e = blockIdx.x * blockDim.x + threadIdx.x;
    float m[KH];
#pragma unroll
    for (int k = 0; k < KH; ++k) m[k] = -3.0e38f;

    if (e < NTOT) {
        int s, d;
        if (e < N_EDGES) { s = erow[e]; d = ecol[e]; }
        else             { s = e - N_EDGES; d = s; }       // self loop
        atomicAdd(&rowcnt[s], 1);
#pragma unroll
        for (int k = 0; k < KH; ++k) {
            float l = ssrc[k * N_NODES + s] + sdst[k * N_NODES + d];
            m[k] = (l >= 0.0f) ? l : NEG_SLOPE * l;
        }
    }
#pragma unroll
    for (int k = 0; k < KH; ++k) {
        float v = m[k];
#pragma unroll
        for (int o = 16; o; o >>= 1) v = fmaxf(v, __shfl_xor(v, o, 32));
        if ((threadIdx.x & 31) == 0) atomicMax(&hmax[k], fenc(v));
    }
}

// ---------------------------------------------------------------------------
// exclusive prefix sum of degrees -> CSR row offsets (single 1024-thread block)
// also clears the counters so k_fill can reuse them as cursors.
// ---------------------------------------------------------------------------
__global__ void __launch_bounds__(1024) k_scan(int* __restrict__ cnt,
                                               int* __restrict__ offs) {
    __shared__ int buf[1024];
    __shared__ int carry;
    if (threadIdx.x == 0) carry = 0;
    __syncthreads();
    for (int base = 0; base < N_NODES; base += 1024) {
        const int i = base + (int)threadIdx.x;
        const int v = (i < N_NODES) ? cnt[i] : 0;
        buf[threadIdx.x] = v;
        __syncthreads();
        for (int off = 1; off < 1024; off <<= 1) {
            int t = (threadIdx.x >= (unsigned)off) ? buf[threadIdx.x - off] : 0;
            __syncthreads();
            buf[threadIdx.x] += t;
            __syncthreads();
        }
        const int incl = buf[threadIdx.x];
        const int c    = carry;
        if (i < N_NODES) { offs[i] = c + incl - v; cnt[i] = 0; }
        __syncthreads();
        if (threadIdx.x == 1023) carry = c + incl;
        __syncthreads();
    }
    if (threadIdx.x == 0) offs[N_NODES] = carry;   // == NTOT
}

// ---------------------------------------------------------------------------
// scatter edge ids into CSR buckets (rowcnt reused as per-row cursor)
// ---------------------------------------------------------------------------
__global__ void k_fill(const int* __restrict__ erow, const int* __restrict__ offs,
                       int* __restrict__ cursor, int* __restrict__ elist) {
    const int e = blockIdx.x * blockDim.x + threadIdx.x;
    if (e >= NTOT) return;
    const int s = (e < N_EDGES) ? erow[e] : e - N_EDGES;
    const int p = offs[s] + atomicAdd(&cursor[s], 1);
    elist[p] = e;
}

// ---------------------------------------------------------------------------
// aggregation: one wave per node. 4 heads accumulated in registers
// (4 x v4f per lane), exp-softmax weights, gate by e[n][k], residual +x.
// h gathers are 128-bit loads from L2-resident h (102 MB < 192 MB L2).
// ---------------------------------------------------------------------------
__global__ void __launch_bounds__(256) k_agg(const float* __restrict__ x,
                                             const float* __restrict__ egate,
                                             const float* __restrict__ h,
                                             const float* __restrict__ ssrc,
                                             const float* __restrict__ sdst,
                                             const int* __restrict__ ecol,
                                             const unsigned* __restrict__ hmax,
                                             const int* __restrict__ offs,
                                             const int* __restrict__ elist,
                                             float* __restrict__ out) {
    const int lane = threadIdx.x & 31;
    const int wave = threadIdx.x >> 5;
    const int n    = blockIdx.x * 8 + wave;
    if (n >= N_NODES) return;

    float mx[KH], ss[KH], den[KH];
    v4f acc[KH];
#pragma unroll
    for (int k = 0; k < KH; ++k) {
        mx[k]  = fdec(hmax[k]);
        ss[k]  = ssrc[k * N_NODES + n];
        den[k] = 0.0f;
        acc[k] = (v4f){0.0f, 0.0f, 0.0f, 0.0f};
    }

    const int jb = offs[n], je = offs[n + 1];
    for (int j = jb; j < je; ++j) {
        const int eid = elist[j];
        const int d   = (eid < N_EDGES) ? ecol[eid] : eid - N_EDGES;
#pragma unroll
        for (int k = 0; k < KH; ++k) {
            float l = ss[k] + sdst[k * N_NODES + d];
            l = (l >= 0.0f) ? l : NEG_SLOPE * l;
            const float w = expf(l - mx[k]);
            den[k] += w;
            const v4f hv = *(const v4f*)(h + ((size_t)k * N_NODES + d) * OUT_F + lane * 4);
            acc[k] += w * hv;
        }
    }

    v4f r = *(const v4f*)(x + (size_t)n * IN_F + lane * 4);
#pragma unroll
    for (int k = 0; k < KH; ++k) {
        const float g = egate[(size_t)n * KH + k] / (den[k] + EPS_F);
        r += g * acc[k];
    }
    *(v4f*)(out + (size_t)n * OUT_F + lane * 4) = r;
}

// ---------------------------------------------------------------------------
extern "C" void kernel_launch(void* const* d_in, const int* in_sizes, int n_in,
                              void* d_out, int out_size, void* d_ws, size_t ws_size,
                              hipStream_t stream) {
    (void)in_sizes; (void)n_in; (void)out_size; (void)ws_size;

    const float* x  = (const float*)d_in[0];   // [N_NODES][IN_F]
    const float* eg = (const float*)d_in[1];   // [N_NODES][KH]
    const float* W  = (const float*)d_in[2];   // [KH][IN_F][OUT_F]
    const float* a  = (const float*)d_in[3];   // [KH][2*OUT_F]
    const int*  adj = (const int*)d_in[4];     // [2][N_EDGES] (int32)
    const int* erow = adj;
    const int* ecol = adj + N_EDGES;
    float* out = (float*)d_out;

    // workspace layout (bytes): ~108.6 MB total
    char* w = (char*)d_ws;
    float*    h      = (float*)(w);                   // 102,400,000
    float*    ssrc   = (float*)(w + 102400000);       //     800,000
    float*    sdst   = (float*)(w + 103200000);       //     800,000
    unsigned* hmax   = (unsigned*)(w + 104000000);    //         256 (4 used)
    int*      offs   = (int*)(w + 104000256);         //     200,004 -> pad
    int*      rowcnt = (int*)(w + 104200448);         //     200,000 -> pad
    int*      elist  = (int*)(w + 104400448);         //   4,200,000

    k_init  <<<(N_NODES + 255) / 256, 256, 0, stream>>>(hmax, rowcnt);
    k_gemm  <<<dim3(N_NODES / 16, KH), 256, 0, stream>>>(x, W, h);
    k_scores<<<(KH * N_NODES + 7) / 8, 256, 0, stream>>>(h, a, ssrc, sdst);
    k_edge  <<<(NTOT + 255) / 256, 256, 0, stream>>>(erow, ecol, ssrc, sdst, hmax, rowcnt);
    k_scan  <<<1, 1024, 0, stream>>>(rowcnt, offs);
    k_fill  <<<(NTOT + 255) / 256, 256, 0, stream>>>(erow, offs, rowcnt, elist);
    k_agg   <<<(N_NODES + 7) / 8, 256, 0, stream>>>(x, eg, h, ssrc, sdst, ecol,
                                                    hmax, offs, elist, out);
}